// MultiClassNMS_63943473103309
// MI455X (gfx1250) — compile-verified
//
#include <hip/hip_runtime.h>
#include <hip/hip_bf16.h>
#include <math.h>

// ---------------- problem constants (from reference) ----------------
#define BATCH    8
#define NBOX     16384
#define NCLS     80
#define TDET     100
#define SCORE_TH 0.3f
#define IOU_TH   0.5f

#define NTHREADS   256
#define PER_THREAD (NBOX / NTHREADS)   // 64 scores per thread, held in VGPRs
#define CAND       (NCLS * TDET)       // 8000 per-class detections per batch
#define NEG_INF    (-INFINITY)

// CDNA5 async global->LDS path (ASYNCcnt-tracked), guarded so compile never fails.
#if defined(__HIP_DEVICE_COMPILE__)
#  if __has_builtin(__builtin_amdgcn_global_load_async_to_lds_b128) && \
      __has_builtin(__builtin_amdgcn_s_wait_asynccnt)
#    define USE_ASYNC_LDS 1
#  else
#    define USE_ASYNC_LDS 0
#  endif
#  if __has_builtin(__builtin_amdgcn_s_wait_tensorcnt)
#    define HAVE_WAIT_TENSOR 1
#  else
#    define HAVE_WAIT_TENSOR 0
#  endif
#else
#  define USE_ASYNC_LDS 0
#  define HAVE_WAIT_TENSOR 0
#endif

// Builtin signature (from hipcc diagnostic):
//   void __builtin_amdgcn_global_load_async_to_lds_b128(int4 AS1* src, int4 AS3* dst,
//                                                       imm int offset, imm int cpol)
typedef int v4i __attribute__((ext_vector_type(4)));
typedef __attribute__((address_space(1))) v4i gv4i;   // global int4
typedef __attribute__((address_space(3))) v4i lv4i;   // LDS int4

// --------------------------------------------------------------------------
// Kernel 1: greedy NMS for one (batch, class) pair per block.
//   - boxes of the batch staged into LDS (256 KB) via async global->LDS copies
//   - per-thread scores live in 64 VGPRs, suppressed in place
//   - serial loop of 100 steps: block-wide argmax -> broadcast -> IoU suppress
// --------------------------------------------------------------------------
__global__ __launch_bounds__(NTHREADS) void nms_class_kernel(
    const float* __restrict__ boxes,      // (B, N, 4)
    const float* __restrict__ scores,     // (B, N, C)
    float* __restrict__ det_boxes,        // (B, C, T, 4)
    float* __restrict__ det_scores,       // (B, C, T)
    int*   __restrict__ det_kept)         // (B, C, T)
{
    extern __shared__ char smem[];
    float* lds_boxes = (float*)smem;                                   // NBOX*4 floats
    float* rv = (float*)(smem + (size_t)NBOX * 4 * sizeof(float));     // per-wave max val
    int*   ri = (int*)(rv + 8);                                        // per-wave max idx
    float* bc = (float*)(ri + 8);  // broadcast: [0]=val [1]=idx-bits [2..5]=box [6]=area

    const int tid = threadIdx.x;
    const int b   = blockIdx.x / NCLS;
    const int c   = blockIdx.x % NCLS;

    const float* batch_boxes = boxes + (size_t)b * NBOX * 4;
    const float* sc_base     = scores + (size_t)b * NBOX * NCLS + c;

    // hint the strided score gather while we stage boxes
    __builtin_prefetch(sc_base + (size_t)tid * NCLS, 0, 0);

    // ---- stage all boxes of this batch into LDS ----
#if USE_ASYNC_LDS
    for (int k = 0; k < PER_THREAD; ++k) {
        const int n = tid + k * NTHREADS;
        __builtin_amdgcn_global_load_async_to_lds_b128(
            (gv4i*)(void*)(batch_boxes + (size_t)n * 4),
            (lv4i*)(void*)(lds_boxes + (size_t)n * 4),
            0, 0);
    }
    __builtin_amdgcn_s_wait_asynccnt(0);
#else
    for (int k = 0; k < PER_THREAD; ++k) {
        const int n = tid + k * NTHREADS;
        float4 v = *(const float4*)(batch_boxes + (size_t)n * 4);
        *(float4*)(lds_boxes + (size_t)n * 4) = v;
    }
#endif
#if HAVE_WAIT_TENSOR
    __builtin_amdgcn_s_wait_tensorcnt(0);   // gfx1250 split-counter wait (trivially satisfied)
#endif
    __syncthreads();

    // ---- load this class's scores into registers, apply score threshold ----
    float sv[PER_THREAD];
#pragma unroll
    for (int k = 0; k < PER_THREAD; ++k) {
        const int n = tid + k * NTHREADS;
        const float s = sc_base[(size_t)n * NCLS];
        sv[k] = (s > SCORE_TH) ? s : NEG_INF;
    }

    const size_t obase = ((size_t)b * NCLS + c) * TDET;

    for (int t = 0; t < TDET; ++t) {
        // ---- local argmax over this thread's 64 scores (first-index tie-break) ----
        float bestv = NEG_INF;
        int   besti = 0x7FFFFFFF;
#pragma unroll
        for (int k = 0; k < PER_THREAD; ++k) {
            const int n = tid + k * NTHREADS;
            const float v = sv[k];
            if (v > bestv || (v == bestv && n < besti)) { bestv = v; besti = n; }
        }
        // ---- wave32 reduce ----
        for (int off = 16; off > 0; off >>= 1) {
            const float ov = __shfl_xor(bestv, off, 32);
            const int   oi = __shfl_xor(besti, off, 32);
            if (ov > bestv || (ov == bestv && oi < besti)) { bestv = ov; besti = oi; }
        }
        if ((tid & 31) == 0) { rv[tid >> 5] = bestv; ri[tid >> 5] = besti; }
        __syncthreads();
        // ---- cross-wave reduce + broadcast selected box ----
        if (tid == 0) {
            float bv = rv[0]; int bi = ri[0];
            for (int w = 1; w < NTHREADS / 32; ++w) {
                const float ov = rv[w]; const int oi = ri[w];
                if (ov > bv || (ov == bv && oi < bi)) { bv = ov; bi = oi; }
            }
            const float4 sb = ((const float4*)lds_boxes)[bi];
            bc[0] = bv; ((int*)bc)[1] = bi;
            bc[2] = sb.x; bc[3] = sb.y; bc[4] = sb.z; bc[5] = sb.w;
            bc[6] = fmaxf(sb.z - sb.x, 0.f) * fmaxf(sb.w - sb.y, 0.f);
        }
        __syncthreads();
        const float bv   = bc[0];
        const int   bi   = ((int*)bc)[1];
        const float sy1  = bc[2], sx1 = bc[3], sy2 = bc[4], sx2 = bc[5];
        const float asel = bc[6];

        if (!(bv > NEG_INF)) {  // class exhausted: zero-fill remaining slots, done
            for (int tt = t + tid; tt < TDET; tt += NTHREADS) {
                det_scores[obase + tt] = 0.f;
                det_kept[obase + tt]   = 0;
                float4 z = {0.f, 0.f, 0.f, 0.f};
                ((float4*)det_boxes)[obase + tt] = z;
            }
            break;
        }
        if (tid == 0) {
            det_scores[obase + t] = bv;
            det_kept[obase + t]   = 1;
            float4 sb = {sy1, sx1, sy2, sx2};
            ((float4*)det_boxes)[obase + t] = sb;
        }
        // ---- IoU suppression against the selected box (boxes from LDS) ----
#pragma unroll
        for (int k = 0; k < PER_THREAD; ++k) {
            const int n = tid + k * NTHREADS;
            const float4 o = ((const float4*)lds_boxes)[n];
            const float yy1 = fmaxf(sy1, o.x);
            const float xx1 = fmaxf(sx1, o.y);
            const float yy2 = fminf(sy2, o.z);
            const float xx2 = fminf(sx2, o.w);
            const float inter = fmaxf(yy2 - yy1, 0.f) * fmaxf(xx2 - xx1, 0.f);
            const float ao    = fmaxf(o.z - o.x, 0.f) * fmaxf(o.w - o.y, 0.f);
            const float un    = asel + ao - inter;
            const float iou   = inter / fmaxf(un, 1e-9f);
            if (iou >= IOU_TH || n == bi) sv[k] = NEG_INF;
        }
        __syncthreads();
    }
}

// --------------------------------------------------------------------------
// Kernel 2: per-batch top-100 over the 8000 class-level detections.
// Iterative argmax with removal over an LDS-resident masked score array
// reproduces top_k's value-desc / index-asc ordering.
// --------------------------------------------------------------------------
__global__ __launch_bounds__(NTHREADS) void topk_kernel(
    const float* __restrict__ det_boxes,   // (B, C*T, 4)
    const float* __restrict__ det_scores,  // (B, C*T)
    const int*   __restrict__ det_kept,    // (B, C*T)
    float* __restrict__ out)               // concatenated outputs
{
    __shared__ float ls[CAND];
    __shared__ float rv[8];
    __shared__ int   ri[8];

    const int tid = threadIdx.x;
    const int b   = blockIdx.x;
    const size_t base = (size_t)b * CAND;

    for (int i = tid; i < CAND; i += NTHREADS)
        ls[i] = det_kept[base + i] ? det_scores[base + i] : NEG_INF;
    __syncthreads();

    float* out_boxes  = out;                             // B*100*4 f32
    float* out_scores = out + (size_t)BATCH * TDET * 4;  // B*100   f32
    int*   out_labels = (int*)(out + (size_t)BATCH * TDET * 5);  // B*100 i32
    int*   out_valid  = (int*)(out + (size_t)BATCH * TDET * 6);  // B     i32

    int cnt = 0;
    for (int t = 0; t < TDET; ++t) {
        float bestv = NEG_INF;
        int   besti = 0x7FFFFFFF;
        for (int i = tid; i < CAND; i += NTHREADS) {
            const float v = ls[i];
            if (v > bestv || (v == bestv && i < besti)) { bestv = v; besti = i; }
        }
        for (int off = 16; off > 0; off >>= 1) {
            const float ov = __shfl_xor(bestv, off, 32);
            const int   oi = __shfl_xor(besti, off, 32);
            if (ov > bestv || (ov == bestv && oi < besti)) { bestv = ov; besti = oi; }
        }
        if ((tid & 31) == 0) { rv[tid >> 5] = bestv; ri[tid >> 5] = besti; }
        __syncthreads();
        if (tid == 0) {
            float bv = rv[0]; int bi = ri[0];
            for (int w = 1; w < NTHREADS / 32; ++w) {
                if (rv[w] > bv || (rv[w] == bv && ri[w] < bi)) { bv = rv[w]; bi = ri[w]; }
            }
            const size_t o = (size_t)b * TDET + t;
            if (bv > NEG_INF) {
                ((float4*)out_boxes)[o] = ((const float4*)det_boxes)[base + bi];
                out_scores[o] = bv;
                out_labels[o] = bi / TDET;
                ++cnt;
            } else {
                float4 z = {0.f, 0.f, 0.f, 0.f};
                ((float4*)out_boxes)[o] = z;
                out_scores[o] = 0.f;
                out_labels[o] = 0;
            }
            if (bi >= 0 && bi < CAND) ls[bi] = NEG_INF;  // remove selected
        }
        __syncthreads();
    }
    if (tid == 0) out_valid[b] = cnt;
}

// --------------------------------------------------------------------------
extern "C" void kernel_launch(void* const* d_in, const int* in_sizes, int n_in,
                              void* d_out, int out_size, void* d_ws, size_t ws_size,
                              hipStream_t stream) {
    (void)in_sizes; (void)n_in; (void)out_size; (void)ws_size;

    const float* boxes  = (const float*)d_in[0];   // (B, N, 4) f32
    const float* scores = (const float*)d_in[1];   // (B, N, C) f32

    // workspace layout
    float* det_boxes  = (float*)d_ws;                                        // B*C*T*4
    float* det_scores = det_boxes + (size_t)BATCH * NCLS * TDET * 4;         // B*C*T
    int*   det_kept   = (int*)(det_scores + (size_t)BATCH * NCLS * TDET);    // B*C*T

    const size_t shmem = (size_t)NBOX * 4 * sizeof(float) + 256;  // 256KB boxes + scratch
    (void)hipFuncSetAttribute(reinterpret_cast<const void*>(nms_class_kernel),
                              hipFuncAttributeMaxDynamicSharedMemorySize, (int)shmem);

    nms_class_kernel<<<BATCH * NCLS, NTHREADS, shmem, stream>>>(
        boxes, scores, det_boxes, det_scores, det_kept);

    topk_kernel<<<BATCH, NTHREADS, 0, stream>>>(
        det_boxes, det_scores, det_kept, (float*)d_out);
}